// GNN_59854664237127
// MI455X (gfx1250) — compile-verified
//
#include <hip/hip_runtime.h>
#include <hip/hip_bf16.h>

typedef __attribute__((ext_vector_type(2))) float v2f;
typedef __attribute__((ext_vector_type(8))) float v8f;

#define NNODES 50000
#define NEDGES 800000
#define DH 64

// ---------------------------------------------------------------------------
// Degree / normalization kernels
// ---------------------------------------------------------------------------
__global__ void deg_init_kernel(float* __restrict__ deg, int n) {
    int i = blockIdx.x * blockDim.x + threadIdx.x;
    if (i < n) deg[i] = 1.0f;                 // self-loop contributes 1
}

__global__ void deg_count_kernel(const int* __restrict__ dst, float* __restrict__ deg, int e) {
    int i = blockIdx.x * blockDim.x + threadIdx.x;
    if (i < e) atomicAdd(&deg[dst[i]], 1.0f);
}

__global__ void deg_rsqrt_kernel(float* __restrict__ deg, int n) {
    int i = blockIdx.x * blockDim.x + threadIdx.x;
    if (i < n) deg[i] = rsqrtf(deg[i]);       // deg^{-1/2}
}

// ---------------------------------------------------------------------------
// Full-fp32 WMMA GEMM:  C[M,N] = A[M,K] @ B[K,N] (+ bias[N] if bias != null)
// One wave computes one 16x16 D tile via V_WMMA_F32_16X16X4_F32 over K/4 steps.
// Block = 4 waves covering N = 64; grid.x = M/16.
// Fragment layouts per CDNA5 ISA 7.12.2 (wave32):
//   A 16x4 : lanes 0-15 hold M=lane (K = 0,1 in v0,v1); lanes 16-31 K = 2,3
//   B 4x16 : v0 = rows K+0 (lanes 0-15) / K+2 (lanes 16-31); v1 = K+1 / K+3
//   C/D    : VGPR r = row r (lanes 0-15) / row r+8 (lanes 16-31), N = lane&15
// ---------------------------------------------------------------------------
__global__ void wmma_gemm_bias_kernel(const float* __restrict__ A,
                                      const float* __restrict__ B,
                                      const float* __restrict__ bias,
                                      float* __restrict__ C,
                                      int N, int K) {
    const int lane = threadIdx.x & 31;
    const int wave = threadIdx.x >> 5;
    const int half = lane >> 4;          // 0: lanes 0-15, 1: lanes 16-31
    const int l    = lane & 15;
    const int m0   = blockIdx.x * 16;
    const int n0   = wave * 16;

    const float* arow = A + (size_t)(m0 + l) * K + half * 2;
    const float* bcol = B + (size_t)(half * 2) * N + n0 + l;

    v8f c = {};
    for (int k = 0; k < K; k += 4) {
        v2f a, b;
        a[0] = arow[k];
        a[1] = arow[k + 1];
        b[0] = bcol[(size_t)k * N];
        b[1] = bcol[(size_t)(k + 1) * N];
        c = __builtin_amdgcn_wmma_f32_16x16x4_f32(
                /*neg_a=*/false, a, /*neg_b=*/false, b,
                /*c_mod=*/(short)0, c, /*reuse_a=*/false, /*reuse_b=*/false);
    }

    const float bv = bias ? bias[n0 + l] : 0.0f;
    float* crow = C + (size_t)(m0 + half * 8) * N + n0 + l;
#pragma unroll
    for (int r = 0; r < 8; ++r) {
        crow[(size_t)r * N] = c[r] + bv;
    }
}

// ---------------------------------------------------------------------------
// agg[i,j] = hw[i,j] * dinv[i]^2 + bias[j]   (self-loop message + bias)
// ---------------------------------------------------------------------------
__global__ void init_agg_kernel(const float* __restrict__ hw,
                                const float* __restrict__ dinv,
                                const float* __restrict__ bias,
                                float* __restrict__ agg, int n) {
    int idx = blockIdx.x * blockDim.x + threadIdx.x;   // n * 64 threads
    if (idx >= n * DH) return;
    int i = idx >> 6;
    int j = idx & (DH - 1);
    float di = dinv[i];
    agg[idx] = hw[idx] * (di * di) + bias[j];
}

// ---------------------------------------------------------------------------
// Edge scatter: agg[dst] += hw[src] * dinv[src]*dinv[dst]
// 16 threads per edge, float4 gather + 4 f32 atomics (resolve in L2: agg is
// 12.8 MB, far under the 192 MB global L2).
// ---------------------------------------------------------------------------
__global__ void scatter_kernel(const float* __restrict__ hw,
                               const int* __restrict__ src,
                               const int* __restrict__ dst,
                               const float* __restrict__ dinv,
                               float* __restrict__ agg, int e) {
    int idx = blockIdx.x * blockDim.x + threadIdx.x;   // e * 16 threads
    int ei = idx >> 4;
    if (ei >= e) return;
    int j4 = (idx & 15) << 2;
    int s = src[ei];
    int d = dst[ei];
    float norm = dinv[s] * dinv[d];
    const float4 v = *(const float4*)(hw + (size_t)s * DH + j4);
    float* base = agg + (size_t)d * DH + j4;
    atomicAdd(base + 0, v.x * norm);
    atomicAdd(base + 1, v.y * norm);
    atomicAdd(base + 2, v.z * norm);
    atomicAdd(base + 3, v.w * norm);
}

__global__ void relu_copy_kernel(const float* __restrict__ in,
                                 float* __restrict__ out, int n) {
    int i = blockIdx.x * blockDim.x + threadIdx.x;
    if (i < n) out[i] = fmaxf(in[i], 0.0f);
}

// ---------------------------------------------------------------------------
// Launch
// ---------------------------------------------------------------------------
extern "C" void kernel_launch(void* const* d_in, const int* in_sizes, int n_in,
                              void* d_out, int out_size, void* d_ws, size_t ws_size,
                              hipStream_t stream) {
    const float* x     = (const float*)d_in[0];   // [N, 128]
    const int*   ei    = (const int*)  d_in[1];   // [2, E]
    const float* lin_w = (const float*)d_in[2];   // [128, 64]
    const float* lin_b = (const float*)d_in[3];   // [64]
    const float* cw1   = (const float*)d_in[4];
    const float* cb1   = (const float*)d_in[5];
    const float* cw2   = (const float*)d_in[6];
    const float* cb2   = (const float*)d_in[7];
    const float* cw3   = (const float*)d_in[8];
    const float* cb3   = (const float*)d_in[9];
    float* out = (float*)d_out;

    const int N = NNODES, E = NEDGES;
    const int* src = ei;
    const int* dst = ei + E;

    float* h    = (float*)d_ws;                  // N*64
    float* hw   = h   + (size_t)N * DH;          // N*64
    float* agg  = hw  + (size_t)N * DH;          // N*64
    float* dinv = agg + (size_t)N * DH;          // N

    const int feat  = N * DH;                    // 3.2M
    const int fblk  = (feat + 255) / 256;
    const int nblk  = (N + 255) / 256;
    const int eblk  = (E + 255) / 256;
    const int sblk  = (E * 16 + 255) / 256;      // 12.8M threads
    const int mtile = N / 16;                    // 3125, exact

    // --- degree / normalization ---
    deg_init_kernel<<<nblk, 256, 0, stream>>>(dinv, N);
    deg_count_kernel<<<eblk, 256, 0, stream>>>(dst, dinv, E);
    deg_rsqrt_kernel<<<nblk, 256, 0, stream>>>(dinv, N);

    // --- linear: h = x @ lin_w + lin_b ---
    wmma_gemm_bias_kernel<<<mtile, 128, 0, stream>>>(x, lin_w, lin_b, h, DH, 128);

    // --- conv 1 (+ReLU) ---
    wmma_gemm_bias_kernel<<<mtile, 128, 0, stream>>>(h, cw1, nullptr, hw, DH, DH);
    init_agg_kernel<<<fblk, 256, 0, stream>>>(hw, dinv, cb1, agg, N);
    scatter_kernel<<<sblk, 256, 0, stream>>>(hw, src, dst, dinv, agg, E);
    relu_copy_kernel<<<fblk, 256, 0, stream>>>(agg, h, feat);

    // --- conv 2 (+ReLU) ---
    wmma_gemm_bias_kernel<<<mtile, 128, 0, stream>>>(h, cw2, nullptr, hw, DH, DH);
    init_agg_kernel<<<fblk, 256, 0, stream>>>(hw, dinv, cb2, agg, N);
    scatter_kernel<<<sblk, 256, 0, stream>>>(hw, src, dst, dinv, agg, E);
    relu_copy_kernel<<<fblk, 256, 0, stream>>>(agg, h, feat);

    // --- conv 3 (no ReLU) -> d_out directly ---
    wmma_gemm_bias_kernel<<<mtile, 128, 0, stream>>>(h, cw3, nullptr, hw, DH, DH);
    init_agg_kernel<<<fblk, 256, 0, stream>>>(hw, dinv, cb3, out, N);
    scatter_kernel<<<sblk, 256, 0, stream>>>(hw, src, dst, dinv, out, E);
}